// S4_layer_72937134621216
// MI455X (gfx1250) — compile-verified
//
#include <hip/hip_runtime.h>
#include <hip/hip_bf16.h>
#include <math.h>
#include <stdint.h>

// ---------------------------------------------------------------------------
// S4(D) layer for MI455X (gfx1250, wave32).
//   u = LayerNorm_H(x)
//   y = gelu( bidir_SSM_conv(u) + D*u )        (exact chunked scan, T=32)
//   z = Wout @ y + bout ; out = z[:H] * sigmoid(z[H:])   (WMMA f32 16x16x4,
//       B-tiles staged into LDS by the Tensor Data Mover, double-buffered)
// ---------------------------------------------------------------------------

typedef __attribute__((ext_vector_type(2))) float v2f;
typedef __attribute__((ext_vector_type(8))) float v8f;
typedef unsigned int v4u __attribute__((ext_vector_type(4)));
typedef int v4i __attribute__((ext_vector_type(4)));
typedef int v8i __attribute__((ext_vector_type(8)));

#define DIMH   512
#define STATE  64
#define BSZ    8
#define SEQ    4096
#define CH     32              // scan chunk length
#define NCHUNK (SEQ / CH)      // 128

#if defined(__has_builtin)
#  if __has_builtin(__builtin_amdgcn_tensor_load_to_lds)
#    define HAVE_TDM 1
#  endif
#endif
#ifndef HAVE_TDM
#  define HAVE_TDM 0
#endif

// ============================================================================
// Kernel 1: per-h SSM tables.
//  wpow[h][n][j] = w_n^j (j=0..32), kt[c][h][n] = C*B*(w-1)/A, k01[c][h][t] =
//  2*Re(sum_n kt w^t)  (real kernel prefix used for intra-chunk Toeplitz conv)
// ============================================================================
__global__ __launch_bounds__(64) void s4_tables(
    const float* __restrict__ log_dt, const float* __restrict__ log_A_real,
    const float* __restrict__ A_imag, const float* __restrict__ Bp,
    const float* __restrict__ C_re,   const float* __restrict__ C_im,
    float2* __restrict__ wpow_g, float2* __restrict__ kt_g,
    float* __restrict__ k01_g)
{
    __shared__ float2 wp[STATE][CH + 1];
    __shared__ float2 ktld[2][STATE];
    const int h = blockIdx.x;
    const int n = threadIdx.x;          // 0..63

    const float dt = expf(log_dt[h]);
    const float Ar = -expf(log_A_real[h * STATE + n]);
    const float Ai = A_imag[h * STATE + n];
    const float er = expf(dt * Ar);
    const float wr = er * cosf(dt * Ai);
    const float wi = er * sinf(dt * Ai);
    // g = (w - 1) / A   (complex divide via conjugate)
    const float iden = 1.f / (Ar * Ar + Ai * Ai);
    const float nr = wr - 1.f, ni = wi;
    const float gr = (nr * Ar + ni * Ai) * iden;
    const float gi = (ni * Ar - nr * Ai) * iden;
    const float bb = Bp[h * STATE + n];
#pragma unroll
    for (int c = 0; c < 2; ++c) {
        const float cr = C_re[(c * DIMH + h) * STATE + n];
        const float ci = C_im[(c * DIMH + h) * STATE + n];
        float2 kt;
        kt.x = bb * (cr * gr - ci * gi);
        kt.y = bb * (cr * gi + ci * gr);
        ktld[c][n] = kt;
        kt_g[(c * DIMH + h) * STATE + n] = kt;
    }
    // powers w^0 .. w^32
    float pr = 1.f, pi = 0.f;
    for (int j = 0; j <= CH; ++j) {
        float2 w2; w2.x = pr; w2.y = pi;
        wp[n][j] = w2;
        wpow_g[(h * STATE + n) * (CH + 1) + j] = w2;
        const float tr = pr * wr - pi * wi;
        pi = pr * wi + pi * wr;
        pr = tr;
    }
    __syncthreads();
    // real kernel prefixes K0[t], K1[t], t = 0..31
    const int c = n >> 5;
    const int t = n & 31;
    float s = 0.f;
    for (int n2 = 0; n2 < STATE; ++n2) {
        float2 w2 = wp[n2][t];
        float2 kt = ktld[c][n2];
        s += kt.x * w2.x - kt.y * w2.y;
    }
    k01_g[(c * DIMH + h) * CH + t] = 2.f * s;
}

// ============================================================================
// Kernel 2: LayerNorm over channels H for each (b,l). 256 threads = 8 rows x
// 32 l-columns per pass, coalesced along L; values cached in VGPRs (unrolled).
// ============================================================================
__global__ __launch_bounds__(256) void s4_layernorm(
    const float* __restrict__ x, const float* __restrict__ ln_w,
    const float* __restrict__ ln_b, float* __restrict__ u)
{
    __shared__ float red_s[8][32], red_q[8][32];
    __shared__ float muld[32], rsld[32];
    const int b = blockIdx.y;
    const int l0 = blockIdx.x * 32;
    const int lane_l = threadIdx.x & 31;
    const int rg = threadIdx.x >> 5;          // 0..7

    float cache[64];
    float s = 0.f, q = 0.f;
    const float* xb = x + (size_t)b * DIMH * SEQ + l0 + lane_l;
#pragma unroll
    for (int i = 0; i < 64; ++i) {
        const int hh = rg + i * 8;
        const float v = xb[(size_t)hh * SEQ];
        cache[i] = v; s += v; q += v * v;
    }
    red_s[rg][lane_l] = s; red_q[rg][lane_l] = q;
    __syncthreads();
    if (rg == 0) {
        float ts = 0.f, tq = 0.f;
#pragma unroll
        for (int g = 0; g < 8; ++g) { ts += red_s[g][lane_l]; tq += red_q[g][lane_l]; }
        const float mu = ts * (1.f / DIMH);
        const float var = tq * (1.f / DIMH) - mu * mu;
        muld[lane_l] = mu;
        rsld[lane_l] = rsqrtf(var + 1e-5f);
    }
    __syncthreads();
    const float mu = muld[lane_l], rs = rsld[lane_l];
    float* uo = u + (size_t)b * DIMH * SEQ + l0 + lane_l;
#pragma unroll
    for (int i = 0; i < 64; ++i) {
        const int hh = rg + i * 8;
        uo[(size_t)hh * SEQ] = (cache[i] - mu) * rs * ln_w[hh] + ln_b[hh];
    }
}

// ============================================================================
// Kernel 3: bidirectional chunked diagonal-SSM scan, one wave32 per (b,h).
// ============================================================================
__device__ __forceinline__ float gelu_exact(float v) {
    return 0.5f * v * (1.f + erff(v * 0.70710678118654752f));
}

__global__ __launch_bounds__(32) void s4_scan(
    const float* __restrict__ u, const float2* __restrict__ wpow_g,
    const float2* __restrict__ kt_g, const float* __restrict__ k01_g,
    const float* __restrict__ Dp, float* __restrict__ y)
{
    __shared__ float2 wld[STATE * (CH + 1)];   // w_n^j, n-major
    __shared__ float2 ktl[2][STATE];
    __shared__ float  kl[2][CH];
    __shared__ float  uld[CH];
    __shared__ float  sre[STATE], sim[STATE];

    const int bh = blockIdx.x;
    const int b = bh >> 9, h = bh & (DIMH - 1);
    const int tid = threadIdx.x;               // 0..31

    for (int idx = tid; idx < STATE * (CH + 1); idx += 32)
        wld[idx] = wpow_g[(size_t)h * STATE * (CH + 1) + idx];
    for (int idx = tid; idx < 2 * STATE; idx += 32) {
        const int c = idx >> 6, n = idx & (STATE - 1);
        ktl[c][n] = kt_g[(c * DIMH + h) * STATE + n];
    }
    for (int idx = tid; idx < 2 * CH; idx += 32) {
        const int c = idx >> 5, t = idx & (CH - 1);
        kl[c][t] = k01_g[(c * DIMH + h) * CH + t];
    }
    __syncthreads();

    const float* ub = u + (size_t)(b * DIMH + h) * SEQ;
    float* yb = y + (size_t)(b * DIMH + h) * SEQ;
    const float Dh = Dp[h];

    // ---- forward pass ----
    float s0r = 0.f, s0i = 0.f, s1r = 0.f, s1i = 0.f;  // states n=tid, tid+32
    for (int c = 0; c < NCHUNK; ++c) {
        const int p = c * CH;
        uld[tid] = ub[p + tid];
        sre[tid] = s0r; sim[tid] = s0i;
        sre[tid + 32] = s1r; sim[tid + 32] = s1i;
        __syncthreads();
        float acc = 0.f;
        for (int m = 0; m <= tid; ++m) acc += kl[0][tid - m] * uld[m];
#pragma unroll 8
        for (int n = 0; n < STATE; ++n) {
            const float2 w = wld[n * (CH + 1) + tid + 1];
            const float2 kt = ktl[0][n];
            const float er2 = kt.x * w.x - kt.y * w.y;
            const float ei2 = kt.x * w.y + kt.y * w.x;
            acc += 2.f * (er2 * sre[n] - ei2 * sim[n]);
        }
        yb[p + tid] = acc;
        float a0r = 0.f, a0i = 0.f, a1r = 0.f, a1i = 0.f;
#pragma unroll 8
        for (int j = 0; j < CH; ++j) {
            const float uj = uld[j];
            const float2 w0 = wld[tid * (CH + 1) + (CH - 1 - j)];
            a0r += w0.x * uj; a0i += w0.y * uj;
            const float2 w1 = wld[(tid + 32) * (CH + 1) + (CH - 1 - j)];
            a1r += w1.x * uj; a1i += w1.y * uj;
        }
        const float2 wT0 = wld[tid * (CH + 1) + CH];
        const float2 wT1 = wld[(tid + 32) * (CH + 1) + CH];
        float tr = wT0.x * s0r - wT0.y * s0i + a0r;
        s0i = wT0.x * s0i + wT0.y * s0r + a0i; s0r = tr;
        tr = wT1.x * s1r - wT1.y * s1i + a1r;
        s1i = wT1.x * s1i + wT1.y * s1r + a1i; s1r = tr;
        __syncthreads();
    }

    // ---- backward pass + finalize ----
    float r0r = 0.f, r0i = 0.f, r1r = 0.f, r1i = 0.f;
    for (int c = NCHUNK - 1; c >= 0; --c) {
        const int p = c * CH;
        uld[tid] = ub[p + tid];
        sre[tid] = r0r; sim[tid] = r0i;
        sre[tid + 32] = r1r; sim[tid + 32] = r1i;
        __syncthreads();
        float acc = 0.f;
        for (int m = tid + 1; m < CH; ++m) acc += kl[1][m - tid - 1] * uld[m];
#pragma unroll 8
        for (int n = 0; n < STATE; ++n) {
            const float2 w = wld[n * (CH + 1) + (CH - 1 - tid)];
            const float2 kt = ktl[1][n];
            const float er2 = kt.x * w.x - kt.y * w.y;
            const float ei2 = kt.x * w.y + kt.y * w.x;
            acc += 2.f * (er2 * sre[n] - ei2 * sim[n]);
        }
        const float v = yb[p + tid] + acc + Dh * uld[tid];
        yb[p + tid] = gelu_exact(v);
        float a0r = 0.f, a0i = 0.f, a1r = 0.f, a1i = 0.f;
#pragma unroll 8
        for (int j = 0; j < CH; ++j) {
            const float uj = uld[j];
            const float2 w0 = wld[tid * (CH + 1) + j];
            a0r += w0.x * uj; a0i += w0.y * uj;
            const float2 w1 = wld[(tid + 32) * (CH + 1) + j];
            a1r += w1.x * uj; a1i += w1.y * uj;
        }
        const float2 wT0 = wld[tid * (CH + 1) + CH];
        const float2 wT1 = wld[(tid + 32) * (CH + 1) + CH];
        float tr = wT0.x * r0r - wT0.y * r0i + a0r;
        r0i = wT0.x * r0i + wT0.y * r0r + a0i; r0r = tr;
        tr = wT1.x * r1r - wT1.y * r1i + a1r;
        r1i = wT1.x * r1i + wT1.y * r1r + a1i; r1r = tr;
        __syncthreads();
    }
}

// ============================================================================
// Kernel 4: z = Wout @ y + bout, out = z_top * sigmoid(z_bot), fused GLU.
// WMMA f32 16x16x4. A tile (64x32 of Wout) via vector loads; B tile (32x64 of
// y, row stride 4096B*... = SEQ floats) staged by the Tensor Data Mover into a
// double-buffered LDS tile, pipelined one K-chunk ahead (TENSORcnt).
// ============================================================================
#if HAVE_TDM
__device__ __forceinline__ void tdm_load_tile32x64(const float* gsrc,
                                                   unsigned lds_off) {
    // Tensor DMA descriptor (ISA cdna5 ch.8): 2-D fp32 tile, 64 x 32 rows.
    const unsigned long long ga = (unsigned long long)(uintptr_t)gsrc;
    v4u g0;
    g0.x = 1u;                                    // count=1, user descriptor
    g0.y = lds_off;                               // lds_addr (bytes)
    g0.z = (unsigned)ga;                          // global_addr[31:0]
    g0.w = (unsigned)(ga >> 32) | (2u << 30);     // global_addr[56:32] | type=2
    v8i g1;
    g1[0] = (int)(2u << 16);                      // data_size = 4 bytes
    g1[1] = (int)((SEQ & 0xffffu) << 16);         // tensor_dim0 = SEQ  [lo16]
    g1[2] = (int)((SEQ >> 16) | ((DIMH & 0xffffu) << 16)); // dim0 hi | dim1 lo
    g1[3] = (int)((DIMH >> 16) | (64u << 16));    // dim1 hi | tile_dim0 = 64
    g1[4] = 32;                                   // tile_dim1 = 32, tile_dim2=0
    g1[5] = SEQ;                                  // tensor_dim0_stride lo32
    g1[6] = 0;                                    // stride hi | dim1_stride lo
    g1[7] = 0;
    v4i g2 = {0, 0, 0, 0};
    v4i g3 = {0, 0, 0, 0};
#if defined(__clang_major__) && (__clang_major__ >= 23)
    v8i g4 = {0, 0, 0, 0, 0, 0, 0, 0};
    __builtin_amdgcn_tensor_load_to_lds(g0, g1, g2, g3, g4, 0);
#else
    __builtin_amdgcn_tensor_load_to_lds(g0, g1, g2, g3, 0);
#endif
}
#endif

__global__ __launch_bounds__(128) void s4_outproj_glu(
    const float* __restrict__ Wout, const float* __restrict__ bout,
    const float* __restrict__ y, float* __restrict__ out)
{
    __shared__ float Ald[64][33];                       // pad 33 vs bank conflicts
    __shared__ __align__(16) float Bld[2][32][64];      // TDM double buffer
    __shared__ float Zld[32][64];

    const int tid = threadIdx.x;
    const int nt = blockIdx.x * 64;
    const int mt = blockIdx.y * 32;
    const int b = blockIdx.z;
    const int wv = tid >> 5;                            // wave 0..3
    const int lane = tid & 31;
    const int half = lane >> 4;                         // 0 / 1
    const int mfrag = lane & 15;
    const int arow = wv * 16 + mfrag;

    const float* ybase = y + (size_t)b * DIMH * SEQ;
    v8f acc[4] = {};                                    // 4 n-subtiles of 16

#if HAVE_TDM
    if (wv == 0)                                        // prime the pipeline
        tdm_load_tile32x64(ybase + nt, (unsigned)(uintptr_t)&Bld[0][0][0]);
#endif

    for (int ci = 0; ci < DIMH / 32; ++ci) {
        const int kk = ci * 32;
        // stage A: 64 rows (32 top-half + 32 bottom-half of Wout) x 32 k
#pragma unroll
        for (int i = 0; i < 4; ++i) {
            const int f = tid + i * 128;
            const int row = f >> 3;
            const int c4 = (f & 7) << 2;
            const int rowg = (row < 32) ? (mt + row) : (DIMH + mt + (row - 32));
            const float4 v = *(const float4*)(Wout + (size_t)rowg * DIMH + kk + c4);
            Ald[row][c4 + 0] = v.x; Ald[row][c4 + 1] = v.y;
            Ald[row][c4 + 2] = v.z; Ald[row][c4 + 3] = v.w;
        }
#if HAVE_TDM
        if (wv == 0) {
            if (ci + 1 < DIMH / 32) {                   // issue next chunk's DMA
                tdm_load_tile32x64(ybase + (size_t)(kk + 32) * SEQ + nt,
                                   (unsigned)(uintptr_t)&Bld[(ci + 1) & 1][0][0]);
                __builtin_amdgcn_s_wait_tensorcnt(1);   // current tile landed
            } else {
                __builtin_amdgcn_s_wait_tensorcnt(0);
            }
        }
#else
        // fallback: stage B with vector loads
#pragma unroll
        for (int i = 0; i < 4; ++i) {
            const int f = tid + i * 128;
            const int row = f >> 4;
            const int c4 = (f & 15) << 2;
            const float4 v = *(const float4*)(ybase + (size_t)(kk + row) * SEQ + nt + c4);
            *(float4*)&Bld[ci & 1][row][c4] = v;
        }
#endif
        if (kk + 32 < DIMH)                             // warm Wout (global_prefetch_b8)
            __builtin_prefetch(Wout + (size_t)(mt + (tid >> 2)) * DIMH + kk + 32 + ((tid & 3) << 3), 0, 3);
        __syncthreads();

        const float (*Bc)[64] = Bld[ci & 1];
#pragma unroll
        for (int k = 0; k < 32; k += 4) {
            v2f a;
            a.x = Ald[arow][k + half * 2];
            a.y = Ald[arow][k + half * 2 + 1];
#pragma unroll
            for (int s2 = 0; s2 < 4; ++s2) {
                v2f bf;
                bf.x = Bc[k + half * 2][s2 * 16 + mfrag];
                bf.y = Bc[k + half * 2 + 1][s2 * 16 + mfrag];
                acc[s2] = __builtin_amdgcn_wmma_f32_16x16x4_f32(
                    false, a, false, bf, (short)0, acc[s2], false, false);
            }
        }
        __syncthreads();
    }

    // epilogue: bias + GLU (bottom waves publish gate half through LDS)
    if (wv >= 2) {
#pragma unroll
        for (int s2 = 0; s2 < 4; ++s2)
#pragma unroll
            for (int v = 0; v < 8; ++v) {
                const int r32 = ((wv & 1) << 4) + v + half * 8;
                const int col = s2 * 16 + mfrag;
                Zld[r32][col] = acc[s2][v] + bout[DIMH + mt + r32];
            }
    }
    __syncthreads();
    if (wv < 2) {
#pragma unroll
        for (int s2 = 0; s2 < 4; ++s2)
#pragma unroll
            for (int v = 0; v < 8; ++v) {
                const int r32 = ((wv & 1) << 4) + v + half * 8;
                const int col = s2 * 16 + mfrag;
                const float zt = acc[s2][v] + bout[mt + r32];
                const float zb = Zld[r32][col];
                out[((size_t)b * DIMH + mt + r32) * SEQ + nt + col] =
                    zt / (1.f + expf(-zb));
            }
    }
}

// ============================================================================
extern "C" void kernel_launch(void* const* d_in, const int* in_sizes, int n_in,
                              void* d_out, int out_size, void* d_ws, size_t ws_size,
                              hipStream_t stream) {
    (void)in_sizes; (void)n_in; (void)out_size; (void)ws_size;
    const float* x          = (const float*)d_in[0];
    const float* ln_w       = (const float*)d_in[1];
    const float* ln_b       = (const float*)d_in[2];
    const float* log_dt     = (const float*)d_in[3];
    const float* log_A_real = (const float*)d_in[4];
    const float* A_imag     = (const float*)d_in[5];
    const float* Bp         = (const float*)d_in[6];
    const float* C_re       = (const float*)d_in[7];
    const float* C_im       = (const float*)d_in[8];
    const float* Dp         = (const float*)d_in[9];
    const float* Wout       = (const float*)d_in[10];
    const float* bout       = (const float*)d_in[11];

    // workspace layout (floats): u | y | wpow | kt | k01  (~153 MB total)
    float* ws = (float*)d_ws;
    float*  u_buf = ws;                                   // 16,777,216
    float*  y_buf = ws + (size_t)16777216;                // 16,777,216
    float2* wpow  = (float2*)(ws + (size_t)33554432);     // 512*64*33 float2
    float2* kt    = (float2*)(ws + (size_t)37879808);     // 2*512*64  float2
    float*  k01   = ws + (size_t)38141952;                // 2*512*32  float

    s4_tables<<<dim3(DIMH), dim3(64), 0, stream>>>(
        log_dt, log_A_real, A_imag, Bp, C_re, C_im, wpow, kt, k01);
    s4_layernorm<<<dim3(SEQ / 32, BSZ), dim3(256), 0, stream>>>(
        x, ln_w, ln_b, u_buf);
    s4_scan<<<dim3(BSZ * DIMH), dim3(32), 0, stream>>>(
        u_buf, wpow, kt, k01, Dp, y_buf);
    s4_outproj_glu<<<dim3(SEQ / 64, DIMH / 32, BSZ), dim3(128), 0, stream>>>(
        Wout, bout, y_buf, (float*)d_out);
}